// NUFFTLayerEnergy1D_72430328480255
// MI455X (gfx1250) — compile-verified
//
#include <hip/hip_runtime.h>
#include <hip/hip_bf16.h>

// ---------------------------------------------------------------------------
// NUFFT spectral-energy layer, collapsed to: circulant kernel h[d] + sparse
// Gaussian gridding + one small dense f32 WMMA GEMM for the "local" term.
//   E[b] = 0.5 * ( Gsum_b^T H Gsum_b  -  sum_p g_p^T H g_p )
// with H[m,m'] = h[(m-m') mod M],
//   h[d] = (1/(2*pi*M^2)) * sum_n [ ReS(n) cos(2*pi*n*d/M) - ImS(n) sin(...) ]
//   S(n) = deconv(n)^2 * ( w0*mult0(n) + w1*mult1(n) )
// ---------------------------------------------------------------------------

#define MGRID   1025
#define MPAD    1040          // next multiple of 16
#define NB      32            // batches
#define NPTS    512
#define WRAD    16            // half-window of Gaussian support (fp32-exact)
#define WIN     (2*WRAD + 1)  // 33

__device__ __constant__ float kPI    = 3.14159265358979323846f;
__device__ __constant__ float kTWOPI = 6.28318530717958647692f;

typedef float v2f __attribute__((ext_vector_type(2)));
typedef float v8f __attribute__((ext_vector_type(8)));

// ---- kernel 1: spectral factor ReS/ImS, also zero the accumulators --------
__global__ void spec_kernel(const float* __restrict__ mRe0, const float* __restrict__ mIm0,
                            const float* __restrict__ mRe1, const float* __restrict__ mIm1,
                            const float* __restrict__ w0,   const float* __restrict__ w1,
                            float* __restrict__ ReS, float* __restrict__ ImS,
                            float* __restrict__ localAcc)
{
    int i = blockIdx.x * blockDim.x + threadIdx.x;
    if (i < NB) localAcc[i] = 0.0f;
    if (i >= MGRID) return;
    const float tau = 12.0f / (1025.0f * 1025.0f);
    float n  = (float)(i - MGRID / 2);              // KGRID = integers -512..512
    float d2 = (kPI / tau) * __expf(2.0f * n * n * tau);   // deconv(n)^2
    float wa = w0[0], wb = w1[0];
    ReS[i] = d2 * (wa * mRe0[i] + wb * mRe1[i]);
    ImS[i] = d2 * (wa * mIm0[i] + wb * mIm1[i]);
}

// ---- kernel 2: circulant convolution kernel h[d] --------------------------
__global__ void hker(const float* __restrict__ ReS, const float* __restrict__ ImS,
                     float* __restrict__ h)
{
    int d = blockIdx.x * blockDim.x + threadIdx.x;
    if (d >= MGRID) return;
    const float scale = 1.0f / (6.28318530717958647692f * 1025.0f * 1025.0f);
    float acc = 0.0f;
    for (int i = 0; i < MGRID; ++i) {
        int n = i - MGRID / 2;
        int r = (n * d) % MGRID;                 // exact integer angle reduction
        if (r < 0) r += MGRID;
        float th = (kTWOPI / (float)MGRID) * (float)r;
        float s, c;
        __sincosf(th, &s, &c);
        acc = __fmaf_rn(ReS[i], c, acc);
        acc = __fmaf_rn(-ImS[i], s, acc);
    }
    h[d] = acc * scale;
}

// ---- kernel 3: per-batch gridding, diag quadratic forms, Gsum in LDS ------
__global__ void grid_diag_kernel(const float* __restrict__ x, const float* __restrict__ h,
                                 float* __restrict__ GsumPad, float* __restrict__ diag)
{
    __shared__ float Gs[MPAD];
    __shared__ float hwin[2 * (WIN - 1) + 1];     // offsets -32..32 -> 65
    __shared__ float wbuf[256][WIN];
    __shared__ float red[256];

    const int tid = threadIdx.x;
    const int b   = blockIdx.x;

    for (int i = tid; i < MPAD; i += 256) Gs[i] = 0.0f;
    for (int i = tid; i < 65;   i += 256) {
        int d = i - 32;
        hwin[i] = h[(d + MGRID) % MGRID];
    }
    __syncthreads();

    const float DX     = kTWOPI / (float)MGRID;
    const float INVDX  = (float)MGRID / kTWOPI;
    const float INV4T  = 1025.0f * 1025.0f / 48.0f;   // 1/(4*tau)
    float diag_acc = 0.0f;

    for (int pp = 0; pp < NPTS / 256; ++pp) {
        int p = tid + pp * 256;
        float xv = x[b * NPTS + p];
        float u  = (xv + kPI) * INVDX;
        int   c  = (int)floorf(u + 0.5f);
        // Gaussian window (periodic images handled by unwrapped offset + mod)
        for (int j = 0; j < WIN; ++j) {
            int   jj   = c - WRAD + j;
            float diff = (xv + kPI) - (float)jj * DX;
            float w    = __expf(-diff * diff * INV4T);
            wbuf[tid][j] = w;
            int m = jj % MGRID; if (m < 0) m += MGRID;
            atomicAdd(&Gs[m], w);                 // ds_add_f32
        }
        // diag_p = sum_{a,b} w_a * h[(a-b)] * w_b
        for (int a = 0; a < WIN; ++a) {
            float s = 0.0f;
            for (int q = 0; q < WIN; ++q)
                s = __fmaf_rn(hwin[(a - q) + 32], wbuf[tid][q], s);
            diag_acc = __fmaf_rn(wbuf[tid][a], s, diag_acc);
        }
    }

    red[tid] = diag_acc;
    __syncthreads();
    for (int off = 128; off > 0; off >>= 1) {
        if (tid < off) red[tid] += red[tid + off];
        __syncthreads();
    }
    if (tid == 0) diag[b] = red[0];
    __syncthreads();
    for (int i = tid; i < MPAD; i += 256)
        GsumPad[b * MPAD + i] = Gs[i];            // cols 1025..1039 stay zero
}

// ---- kernel 4: local[b] = Gsum^T H Gsum via f32 WMMA 16x16x4 --------------
// A = GsumPad [32 x 1040] row-major; B operand indexed circulantly from h.
// grid: 65 column tiles; block: 64 threads = 2 waves (row tiles 0-15, 16-31).
__global__ void wmma_local_kernel(const float* __restrict__ A, const float* __restrict__ h,
                                  float* __restrict__ localAcc)
{
    const int lane    = threadIdx.x & 31;
    const int wave    = threadIdx.x >> 5;          // 0/1 -> batch rows 0-15 / 16-31
    const int rowbase = wave * 16;
    const int hi      = lane >> 4;                 // K sub-offset select
    const int lo      = lane & 15;
    const int col     = blockIdx.x * 16 + lo;      // output column n (0..1039)

    // circulant index r = (col - k) mod M, walked down by 4 per K-step
    int r = (col - 2 * hi) % MGRID; if (r < 0) r += MGRID;

    const float* arow = A + (size_t)(rowbase + lo) * MPAD;
    v8f acc = {};
    for (int k0 = 0; k0 < MPAD; k0 += 4) {
        int ka = k0 + 2 * hi;                      // lanes 16-31 take K+2 (ISA A/B layout)
        v2f a, bf;
        a.x = arow[ka];
        a.y = arow[ka + 1];
        int rm1 = r - 1; if (rm1 < 0) rm1 += MGRID;
        bf.x = h[r];                               // B[k][col]   = h[(col-k)   mod M]
        bf.y = h[rm1];                             // B[k+1][col] = h[(col-k-1) mod M]
        acc = __builtin_amdgcn_wmma_f32_16x16x4_f32(
            /*neg_a=*/false, a, /*neg_b=*/false, bf,
            /*c_mod=*/(short)0, acc, /*reuse_a=*/false, /*reuse_b=*/false);
        r -= 4; if (r < 0) r += MGRID;
    }
    // Y[b, col] held per C-layout: vgpr v, lane half -> batch = rowbase + v + 8*hi
    if (col < MGRID) {
#pragma unroll
        for (int v = 0; v < 8; ++v) {
            int bidx = rowbase + v + 8 * hi;
            atomicAdd(&localAcc[bidx], acc[v] * A[(size_t)bidx * MPAD + col]);
        }
    }
}

// ---- kernel 5: E[b] = (local - diag) / 2 ----------------------------------
__global__ void finalize_kernel(const float* __restrict__ localAcc,
                                const float* __restrict__ diag,
                                float* __restrict__ out)
{
    int b = threadIdx.x;
    if (b < NB) out[b] = 0.5f * (localAcc[b] - diag[b]);
}

extern "C" void kernel_launch(void* const* d_in, const int* in_sizes, int n_in,
                              void* d_out, int out_size, void* d_ws, size_t ws_size,
                              hipStream_t stream)
{
    const float* x     = (const float*)d_in[0];   // [32,512]
    const float* mRe0  = (const float*)d_in[1];   // [1025]
    const float* mIm0  = (const float*)d_in[2];
    const float* mRe1  = (const float*)d_in[3];
    const float* mIm1  = (const float*)d_in[4];
    const float* w0    = (const float*)d_in[5];   // scalar
    const float* w1    = (const float*)d_in[6];
    float*       out   = (float*)d_out;           // [32] f32

    // workspace layout (floats), ~146 KB total
    float* ws    = (float*)d_ws;
    float* ReS   = ws;                 // 1056
    float* ImS   = ReS + 1056;         // 1056
    float* h     = ImS + 1056;         // 1056
    float* lcl   = h   + 1056;         // 32
    float* diag  = lcl + 32;           // 32
    float* Gsum  = diag + 32;          // 32*1040
    (void)in_sizes; (void)n_in; (void)out_size; (void)ws_size;

    spec_kernel<<<5, 256, 0, stream>>>(mRe0, mIm0, mRe1, mIm1, w0, w1, ReS, ImS, lcl);
    hker<<<(MGRID + 127) / 128, 128, 0, stream>>>(ReS, ImS, h);
    grid_diag_kernel<<<NB, 256, 0, stream>>>(x, h, Gsum, diag);
    wmma_local_kernel<<<MPAD / 16, 64, 0, stream>>>(Gsum, h, lcl);
    finalize_kernel<<<1, 32, 0, stream>>>(lcl, diag, out);
}